// GPoolNet_25348896981066
// MI455X (gfx1250) — compile-verified
//
#include <hip/hip_runtime.h>

typedef __attribute__((ext_vector_type(2))) float v2f;
typedef __attribute__((ext_vector_type(8))) float v8f;

#define DHID 128

// ---------------- elementwise helpers ----------------

__global__ void k_fill(float* __restrict__ p, float v, int n) {
    int i = blockIdx.x * blockDim.x + threadIdx.x;
    if (i < n) p[i] = v;
}

__global__ void k_zero(float* __restrict__ p, long long n) {
    long long i = (long long)blockIdx.x * blockDim.x + threadIdx.x;
    if (i < n) p[i] = 0.0f;
}

__global__ void k_degree(const int* __restrict__ dst, float* __restrict__ deg, int nE) {
    int i = blockIdx.x * blockDim.x + threadIdx.x;
    if (i < nE) atomicAdd(&deg[dst[i]], 1.0f);
}

__global__ void k_rsqrt(float* __restrict__ p, int n) {
    int i = blockIdx.x * blockDim.x + threadIdx.x;
    if (i < n) p[i] = rsqrtf(p[i]);
}

// out[i] = relu(inv[row] * (acc[i] + hs[i]) + bias[col])   (row = i/128, col = i%128)
__global__ void k_post(const float* __restrict__ hs, const float* __restrict__ acc,
                       const float* __restrict__ inv, const float* __restrict__ bias,
                       float* __restrict__ out, long long total) {
    long long i = (long long)blockIdx.x * blockDim.x + threadIdx.x;
    if (i >= total) return;
    long long row = i >> 7;
    int col = (int)(i & 127);
    float v = inv[row] * (acc[i] + hs[i]) + bias[col];
    out[i] = fmaxf(v, 0.0f);
}

// ---------------- edge gather + scatter-add ----------------
// one wave (32 lanes) per edge; float4 per lane covers the 128-float row.
__global__ void k_edge(const int* __restrict__ src, const int* __restrict__ dst,
                       const float* __restrict__ hs, float* __restrict__ acc, int nE) {
    int wavesPerBlock = blockDim.x >> 5;
    int e = blockIdx.x * wavesPerBlock + (threadIdx.x >> 5);
    if (e >= nE) return;
    int lane = threadIdx.x & 31;
    int s = __builtin_amdgcn_readfirstlane(src[e]);   // wave-uniform scalar
    int d = __builtin_amdgcn_readfirstlane(dst[e]);
    const float4 v = *(const float4*)(hs + (size_t)s * DHID + lane * 4);
    float* ap = acc + (size_t)d * DHID + lane * 4;
    atomicAdd(ap + 0, v.x);
    atomicAdd(ap + 1, v.y);
    atomicAdd(ap + 2, v.z);
    atomicAdd(ap + 3, v.w);
}

// ---------------- fp32 WMMA GEMM ----------------
// out[nRows x NCOLS] = A[nRows x 128] @ W[128 x NCOLS]
// NCOLS is compile time: address math = base VGPR + immediate offsets, no guards
// needed when 16 | NCOLS; for NCOLS=40 loads use a clamped (always-safe) column
// plus a cndmask select — no exec-divergent load blocks.
//
// fp32 WMMA lane layouts (ISA 7.12.2):
//   A 16x4:  lanes 0-15 -> M=lane, VGPR0=K0,VGPR1=K1; lanes 16-31 -> K2,K3
//   B 4x16:  lanes 0-15 -> N=lane, VGPR0=K0,VGPR1=K1; lanes 16-31 -> K2,K3
//   C/D:     VGPR i: lanes 0-15 -> M=i,N=lane; lanes 16-31 -> M=i+8,N=lane-16
template <int NCOLS, bool HAS_SCALE, bool HAS_BIAS>
__global__ void k_gemm_wmma(const float* __restrict__ A, const float* __restrict__ W,
                            const float* __restrict__ rowScale, const float* __restrict__ bias,
                            float* __restrict__ out, int nRows) {
    constexpr int NCT = (NCOLS + 15) / 16;            // column tiles
    constexpr bool FULL = (NCOLS % 16) == 0;

    const int wavesPerBlock = blockDim.x >> 5;
    const int wave = blockIdx.x * wavesPerBlock + (threadIdx.x >> 5);
    const int nTiles = (nRows >> 4) * NCT;
    if (wave >= nTiles) return;          // uniform whole-wave exit: EXEC all-1 for WMMA

    const int tr = wave / NCT;
    const int tc = wave % NCT;
    const int lane = threadIdx.x & 31;
    const int half = lane >> 4;          // 0: K pair {0,1}, 1: K pair {2,3}
    const int l = lane & 15;

    const int m = tr * 16 + l;           // A row this lane feeds
    const int n = tc * 16 + l;           // W col this lane feeds
    const bool nOK = FULL || (n < NCOLS);
    const int nSafe = FULL ? n : (n < NCOLS ? n : NCOLS - 1);

    // A: base + immediate offsets (kk*4 bytes)
    const float* __restrict__ arow = A + (size_t)m * DHID + 2 * half;
    // B: base + immediate offsets (kk*NCOLS*4 bytes), always in-bounds
    const float* __restrict__ bcol = W + (size_t)(2 * half) * NCOLS + nSafe;

    v8f c = {};
#pragma unroll
    for (int kk = 0; kk < DHID; kk += 4) {
        v2f a = *(const v2f*)(arow + kk);             // {K=kk+2h, K=kk+2h+1}
        float b0 = bcol[(size_t)kk * NCOLS];          // K=kk+2h
        float b1 = bcol[(size_t)(kk + 1) * NCOLS];    // K=kk+2h+1
        v2f b;
        b.x = nOK ? b0 : 0.0f;                        // v_cndmask, no branch
        b.y = nOK ? b1 : 0.0f;
        c = __builtin_amdgcn_wmma_f32_16x16x4_f32(
                /*neg_a=*/false, a, /*neg_b=*/false, b,
                /*c_mod=*/(short)0, c, /*reuse_a=*/false, /*reuse_b=*/false);
    }

    const int col = tc * 16 + l;
    const float bv = HAS_BIAS ? (nOK ? bias[nSafe] : 0.0f) : 0.0f;
#pragma unroll
    for (int i = 0; i < 8; ++i) {
        const int row = tr * 16 + i + 8 * half;
        float v = c[i];
        if (HAS_SCALE) v *= rowScale[row];
        if (HAS_BIAS)  v += bv;
        if (FULL || col < NCOLS)
            out[(size_t)row * NCOLS + col] = v;
    }
}

// ---------------- launcher ----------------

extern "C" void kernel_launch(void* const* d_in, const int* in_sizes, int n_in,
                              void* d_out, int out_size, void* d_ws, size_t ws_size,
                              hipStream_t stream) {
    const float* x  = (const float*)d_in[0];
    const int*   ei = (const int*)d_in[1];   // [2][E] int32
    const float* W1 = (const float*)d_in[2];
    const float* b1 = (const float*)d_in[3];
    const float* W2 = (const float*)d_in[4];
    const float* b2 = (const float*)d_in[5];
    const float* Wl = (const float*)d_in[6];
    const float* bl = (const float*)d_in[7];
    float* out = (float*)d_out;

    const int N    = in_sizes[0] / DHID;     // 100000
    const int E    = in_sizes[1] / 2;        // 3200000
    constexpr int NCLS = 40;

    const int* src = ei;
    const int* dst = ei + E;

    // workspace layout: inv | bufA | bufB
    size_t invBytes = (((size_t)N * 4) + 255) & ~(size_t)255;
    float* inv  = (float*)d_ws;
    float* bufA = (float*)((char*)d_ws + invBytes);
    float* bufB = bufA + (size_t)N * DHID;

    const long long feat = (long long)N * DHID;

    const int T = 256;
    const int gN    = (N + T - 1) / T;
    const int gE    = (E + T - 1) / T;
    const int gFeat = (int)((feat + T - 1) / T);
    const int wavesPerBlock = T / 32;
    const int gEdgeWave = (E + wavesPerBlock - 1) / wavesPerBlock;

    const int tilesHid  = (N / 16) * (DHID / 16);
    const int gGemmHid  = (tilesHid + wavesPerBlock - 1) / wavesPerBlock;
    const int tilesCls  = (N / 16) * ((NCLS + 15) / 16);
    const int gGemmCls  = (tilesCls + wavesPerBlock - 1) / wavesPerBlock;

    // degree (self-loop => init 1.0) -> inv_sqrt_deg
    k_fill<<<gN, T, 0, stream>>>(inv, 1.0f, N);
    k_degree<<<gE, T, 0, stream>>>(dst, inv, E);
    k_rsqrt<<<gN, T, 0, stream>>>(inv, N);

    // ---- layer 1 ----
    // hs1 = (X @ W1) * inv[row]   -> bufA
    k_gemm_wmma<DHID, true, false><<<gGemmHid, T, 0, stream>>>(x, W1, inv, nullptr, bufA, N);
    k_zero<<<gFeat, T, 0, stream>>>(bufB, feat);
    k_edge<<<gEdgeWave, T, 0, stream>>>(src, dst, bufA, bufB, E);
    // h1 = relu(inv*(acc + hs1) + b1) -> bufB (in place over acc, elementwise)
    k_post<<<gFeat, T, 0, stream>>>(bufA, bufB, inv, b1, bufB, feat);

    // ---- layer 2 ----
    // hs2 = (h1 @ W2) * inv[row] -> bufA
    k_gemm_wmma<DHID, true, false><<<gGemmHid, T, 0, stream>>>(bufB, W2, inv, nullptr, bufA, N);
    k_zero<<<gFeat, T, 0, stream>>>(bufB, feat);
    k_edge<<<gEdgeWave, T, 0, stream>>>(src, dst, bufA, bufB, E);
    // h2 = relu(inv*(acc + hs2) + b2) -> bufB
    k_post<<<gFeat, T, 0, stream>>>(bufA, bufB, inv, b2, bufB, feat);

    // ---- head: out = h2 @ Wl + bl ----
    k_gemm_wmma<NCLS, false, true><<<gGemmCls, T, 0, stream>>>(bufB, Wl, nullptr, bl, out, N);
}